// ViewTransformer_10264971838192
// MI455X (gfx1250) — compile-verified
//
#include <hip/hip_runtime.h>

// ---------------- problem constants (from reference) ----------------
#define BN 2
#define NN 6
#define ND 41
#define NH 28
#define NW 50
#define NC 64
#define GBX 200
#define GBY 200
#define GBZ 20
#define NPTS (ND * NW)          // 2050 points per (n,h) row

typedef __attribute__((ext_vector_type(4))) unsigned int u32x4;
typedef __attribute__((ext_vector_type(8))) int          i32x8;
typedef __attribute__((ext_vector_type(4))) int          i32x4;

// Native CDNA5 no-return f32 atomic add (STOREcnt-tracked), device scope so
// the RMW executes in the L2 atomic units. Avoids any CAS-loop expansion.
__device__ __forceinline__ void atomic_add_f32(float* p, float v) {
  asm volatile("global_atomic_add_f32 %0, %1, off scope:SCOPE_DEV"
               :: "v"(p), "v"(v)
               : "memory");
}

// ---------------- splat: scatter depth-weighted features into voxel grid ---
// (defined first so the disasm snippet shows the TDM path)
__global__ void __launch_bounds__(256) bev_splat_kernel(
    const float* __restrict__ geom,   // [B,N,D,H,W,3]
    const float* __restrict__ feat,   // [B,N,C,H,W]
    const float* __restrict__ depth,  // [B,N,D,H,W]
    float* __restrict__ voxel,        // [BZ,BY,BX,C] for current batch
    int b) {
  __shared__ float featLDS[NC * NW];  // [c][w] tile, 12.8 KB (TDM destination)
  __shared__ int   linLDS[NPTS];
  __shared__ float wLDS[NPTS];

  const int tid = threadIdx.x;
  const int n = blockIdx.x / NH;
  const int h = blockIdx.x % NH;

  // ---- Tensor Data Mover: DMA 2-D tile (dim0 = W contiguous, dim1 = C with
  // stride H*W) of f32 features into LDS. Issued by wave 0 only. ----
  if (tid < 32) {
    const float* gsrc = feat + (((size_t)b * NN + n) * NC) * (NH * NW)
                             + (size_t)h * NW;
    unsigned long long ga = (unsigned long long)gsrc;
    unsigned lds = (unsigned)(unsigned long long)&featLDS[0];
    // D# group 0: count=1 | lds_addr | global_addr[56:0] | type=2
    u32x4 g0 = { 1u,
                 lds,
                 (unsigned)ga,
                 ((unsigned)(ga >> 32) & 0x01FFFFFFu) | (2u << 30) };
    // D# group 1: data_size=4B; tensor_dim0=50; tensor_dim1=64;
    //             tile_dim0=50; tile_dim1=64; tensor_dim0_stride=H*W=1400
    i32x8 g1 = { (int)(2u << 16),            // wg_mask=0, data_size=2 (4B)
                 (int)((unsigned)NW << 16),  // tensor_dim0 lo16 @ bits 63:48
                 (int)((unsigned)NC << 16),  // dim0 hi16=0 | tensor_dim1 lo16
                 (int)((unsigned)NW << 16),  // dim1 hi16=0 | tile_dim0=50
                 NC,                         // tile_dim1=64 | tile_dim2=0
                 NH * NW,                    // tensor_dim0_stride lo32
                 0, 0 };
    i32x4 g2 = { 0, 0, 0, 0 };
    i32x4 g3 = { 0, 0, 0, 0 };
#if defined(__clang_major__) && (__clang_major__ >= 23)
    i32x8 g4 = { 0, 0, 0, 0, 0, 0, 0, 0 };
    __builtin_amdgcn_tensor_load_to_lds(g0, g1, g2, g3, g4, 0);
#else
    __builtin_amdgcn_tensor_load_to_lds(g0, g1, g2, g3, 0);
#endif
  }

  // ---- Phase 1 (overlaps the TDM): per-point voxel index + weight ----
  for (int idx = tid; idx < NPTS; idx += 256) {
    int d  = idx / NW;
    int wx = idx - d * NW;
    size_t gi = ((((size_t)b * NN + n) * ND + d) * NH + h) * NW + wx;
    __builtin_prefetch(geom + 3 * (gi + (size_t)5 * NH * NW), 0, 0);
    float gx = geom[gi * 3 + 0];
    float gy = geom[gi * 3 + 1];
    float gz = geom[gi * 3 + 2];
    float fx = (gx - (-50.0f)) / 0.5f;
    float fy = (gy - (-50.0f)) / 0.5f;
    float fz = (gz - (-10.0f)) / (10.0f - (-10.0f)) * (float)GBZ;
    int xi = (int)fx;   // trunc toward zero == jnp.trunc + astype(int32)
    int yi = (int)fy;
    int zi = (int)fz;
    bool valid = (xi >= 0) && (xi < GBX) && (yi >= 0) && (yi < GBY) &&
                 (zi >= 0) && (zi < GBZ);
    linLDS[idx] = valid ? ((zi * GBY + yi) * GBX + xi) : -1;
    wLDS[idx]   = depth[gi];
  }

  if (tid < 32) __builtin_amdgcn_s_wait_tensorcnt(0);
  __syncthreads();

  // ---- Phase 2: one wave per point; 32 lanes x 2 channels of f32 atomics --
  const int lane = tid & 31;
  const int wvid = tid >> 5;
  for (int p = wvid; p < NPTS; p += 8) {
    int lin = linLDS[p];          // wave-uniform (broadcast LDS read)
    if (lin < 0) continue;        // uniform branch
    float wt = wLDS[p];
    int wx = p % NW;
    float v0 = featLDS[lane * NW + wx] * wt;          // 32 distinct banks
    float v1 = featLDS[(lane + 32) * NW + wx] * wt;
    float* vp = voxel + (size_t)lin * NC;
    atomic_add_f32(vp + lane, v0);
    atomic_add_f32(vp + lane + 32, v1);
  }
}

// ---------------- max over z + transpose to (C, Y, X) ----------------
__global__ void __launch_bounds__(256) bev_max_kernel(
    const float* __restrict__ voxel,  // [BZ,BY,BX,C]
    float* __restrict__ out,          // [B,C,BY,BX]
    int b) {
  __shared__ float tile[GBX * 65];    // pad 64->65 to avoid bank conflicts
  const int y = blockIdx.x;
  const int t = threadIdx.x;
  const size_t zstride = (size_t)GBY * GBX * NC;

  // c-major: consecutive lanes read consecutive channels (coalesced 256B).
  // Non-temporal (last-use): this 204.8 MB stream is dead after the max,
  // keep L2 free for the next batch's accumulator.
  for (int i = 0; i < (GBX * NC) / 256; ++i) {  // 50 iterations
    int idx = t + 256 * i;
    int c = idx & (NC - 1);
    int x = idx >> 6;
    const float* p = voxel + ((size_t)y * GBX + x) * NC + c;
    float m = __builtin_nontemporal_load(p);
#pragma unroll
    for (int z = 1; z < GBZ; ++z)
      m = fmaxf(m, __builtin_nontemporal_load(p + (size_t)z * zstride));
    tile[x * 65 + c] = m;
  }
  __syncthreads();
  // x-major: coalesced non-temporal writes into (B,C,Y,X)
  for (int i = 0; i < (GBX * NC) / 256; ++i) {
    int idx = t + 256 * i;
    int x = idx % GBX;
    int c = idx / GBX;
    __builtin_nontemporal_store(tile[x * 65 + c],
                                &out[(((size_t)b * NC + c) * GBY + y) * GBX + x]);
  }
}

// ---------------- zero the voxel accumulator (float4 stores) ----------------
__global__ void __launch_bounds__(256) bev_zero_kernel(float4* __restrict__ p,
                                                       long long n4) {
  long long i = (long long)blockIdx.x * 256 + threadIdx.x;
  long long stride = (long long)gridDim.x * 256;
  float4 z; z.x = 0.f; z.y = 0.f; z.z = 0.f; z.w = 0.f;
  for (; i < n4; i += stride) p[i] = z;   // RT policy: keep zeros in L2 for RMW
}

// ---------------- launcher ----------------
extern "C" void kernel_launch(void* const* d_in, const int* in_sizes, int n_in,
                              void* d_out, int out_size, void* d_ws,
                              size_t ws_size, hipStream_t stream) {
  const float* geom  = (const float*)d_in[0];   // (2,6,41,28,50,3)
  const float* feat  = (const float*)d_in[1];   // (2,6,64,28,50)
  const float* depth = (const float*)d_in[2];   // (2,6,41,28,50)
  float* out   = (float*)d_out;                 // (2,64,200,200)
  float* voxel = (float*)d_ws;                  // needs 20*200*200*64*4 = 204.8 MB

  const long long vox_elems = (long long)GBZ * GBY * GBX * NC;  // 51.2 M
  for (int b = 0; b < BN; ++b) {
    bev_zero_kernel<<<4096, 256, 0, stream>>>((float4*)voxel, vox_elems / 4);
    bev_splat_kernel<<<NN * NH, 256, 0, stream>>>(geom, feat, depth, voxel, b);
    bev_max_kernel<<<GBY, 256, 0, stream>>>(voxel, out, b);
  }
}